// ELocalLayer_71966472012423
// MI455X (gfx1250) — compile-verified
//
#include <hip/hip_runtime.h>
#include <cstdint>

typedef __attribute__((ext_vector_type(2))) float v2f;
typedef __attribute__((ext_vector_type(8))) float v8f;

#define NU   16
#define FDIM 128

// ---------------------------------------------------------------------------
// Workspace init: zero segment accumulators, winner[] = -1
// ---------------------------------------------------------------------------
__global__ void init_ws_kernel(float* __restrict__ agg1, float* __restrict__ agg2,
                               int* __restrict__ winner, int nagg, int M) {
    int t = blockIdx.x * blockDim.x + threadIdx.x;
    if (t < nagg) { agg1[t] = 0.0f; agg2[t] = 0.0f; }
    if (t < M)    winner[t] = -1;
}

// ---------------------------------------------------------------------------
// Segment sums: agg[idx[e], f] += h[e, f]   (native f32 global atomics)
// ---------------------------------------------------------------------------
__global__ void segsum_kernel(const float* __restrict__ h1, const int* __restrict__ h1i,
                              const float* __restrict__ h2, const int* __restrict__ h2i,
                              float* __restrict__ agg1, float* __restrict__ agg2,
                              unsigned total) {
    unsigned t = blockIdx.x * blockDim.x + threadIdx.x;
    if (t >= total) return;
    unsigned e = t >> 7;        // FDIM == 128
    unsigned f = t & 127u;
    __hip_atomic_fetch_add(&agg1[(unsigned)h1i[e] * FDIM + f], h1[t],
                           __ATOMIC_RELAXED, __HIP_MEMORY_SCOPE_AGENT);
    __hip_atomic_fetch_add(&agg2[(unsigned)h2i[e] * FDIM + f], h2[t],
                           __ATOMIC_RELAXED, __HIP_MEMORY_SCOPE_AGENT);
}

// ---------------------------------------------------------------------------
// Deterministic scatter winner: max-b wins == last-update-wins
// ---------------------------------------------------------------------------
__global__ void winner_kernel(const int* __restrict__ vidx, int* __restrict__ winner, int B) {
    int b = blockIdx.x * blockDim.x + threadIdx.x;
    if (b < B) atomicMax(&winner[vidx[b]], b);
}

// ---------------------------------------------------------------------------
// Bulk copy V_n -> V_n_out (float4, grid-stride, with prefetch)
// ---------------------------------------------------------------------------
__global__ void copy_vn_kernel(const float4* __restrict__ src, float4* __restrict__ dst,
                               size_t n4) {
    size_t t      = (size_t)blockIdx.x * blockDim.x + threadIdx.x;
    size_t stride = (size_t)gridDim.x * blockDim.x;
    for (; t < n4; t += stride) {
        __builtin_prefetch(&src[t + stride], 0, 1);   // global_prefetch_b8 (speculative-safe)
        dst[t] = src[t];
    }
}

// ---------------------------------------------------------------------------
// Main per-b kernel: one wave32 per batch element.
//   D(16x16) = A(16x128) * Bm(128x16) via 32x V_WMMA_F32_16X16X4_F32
//   Bm columns: 0 = feats (mo + h1_agg), 1 = h2_agg, 2 = mo, 3..15 = 0 (padded
//   to full 16 rows in LDS so B-fragment loads need no exec masking).
//   -> col0 = one_body, col1 = two_body, col2 = overlap
//
//   All 32 A-fragments (global) and 32 B-fragments (LDS) are loaded into
//   registers first (bulk issue, single wait), then the 32 WMMAs fire
//   back-to-back so the matrix pipe isn't stalled on per-step memory latency.
// ---------------------------------------------------------------------------
__global__ __launch_bounds__(32)
void elocal_main_kernel(const float* __restrict__ mo,     // [B,128]
                        const float* __restrict__ V_n,    // [M,16,128]
                        const int*   __restrict__ vidx,   // [B]
                        const float* __restrict__ gram,   // [B,16,16]
                        const float* __restrict__ agg1,   // [B,128]
                        const float* __restrict__ agg2,   // [B,128]
                        const int*   __restrict__ winner, // [M]
                        float* __restrict__ energy_out,   // [B]
                        float* __restrict__ vout)         // [M,16,128]
{
    const int b    = blockIdx.x;
    const int lane = threadIdx.x;        // 0..31, wave32
    const int half = lane >> 4;          // 0: lanes 0-15, 1: lanes 16-31
    const int l16  = lane & 15;

    const int    row = vidx[b];
    const float* A   = V_n + (size_t)row * (NU * FDIM);

    __shared__ __align__(16) float sB[NU][FDIM];  // rows: feats/h2_agg/mo, 3..15 = 0
    __shared__ float sD[NU][NU];
    __shared__ float sden[NU];
    __shared__ float s_d2[NU];
    __shared__ float s_num[NU];

    {
        const float* mo_b = mo   + (size_t)b * FDIM;
        const float* a1_b = agg1 + (size_t)b * FDIM;
        const float* a2_b = agg2 + (size_t)b * FDIM;
        for (int c = lane; c < FDIM; c += 32) {
            float mv = mo_b[c];
            sB[0][c] = mv + a1_b[c];   // feats
            sB[1][c] = a2_b[c];        // h2_agg
            sB[2][c] = mv;             // mo_features
            #pragma unroll
            for (int r = 3; r < NU; ++r) sB[r][c] = 0.0f;  // zero padding rows
        }
    }
    __syncthreads();   // single-wave workgroup -> S_NOP

    // --- bulk-load all fragments, then WMMA burst ---------------------------
    // A frag (16x4 f32): lanes 0-15 row M=l16 hold K={0,1}; lanes 16-31 K={2,3}
    // B frag (4x16 f32): lane col N=l16; VGPR0=K0/K2, VGPR1=K1/K3 (same split)
    const float* arow = A + (size_t)l16 * FDIM + 2 * half;
    const float* brow = &sB[l16][2 * half];

    v2f aF[32];
    v2f bF[32];
    #pragma unroll
    for (int i = 0; i < 32; ++i) {
        aF[i] = *(const v2f*)(arow + 4 * i);   // global_load_b64, issued in bulk
        bF[i] = *(const v2f*)(brow + 4 * i);   // ds_load_b64, unconditional
    }

    v8f acc = {};
    #pragma unroll
    for (int i = 0; i < 32; ++i) {
        acc = __builtin_amdgcn_wmma_f32_16x16x4_f32(
                  /*neg_a=*/false, aF[i], /*neg_b=*/false, bF[i],
                  /*c_mod=*/(short)0, acc, /*reuse_a=*/false, /*reuse_b=*/false);
    }

    // spill D to LDS: VGPR r, lane L -> M = r + 8*half, N = l16
    #pragma unroll
    for (int r = 0; r < 8; ++r)
        sD[r + 8 * half][l16] = acc[r];
    __syncthreads();

    // --- denom = gram[b] @ overlap, local energy weighting ------------------
    if (lane < NU) {
        const float* g = gram + ((size_t)b * NU + lane) * NU;
        float d = 0.0f;
        #pragma unroll
        for (int v = 0; v < NU; ++v) d += g[v] * sD[v][2];   // overlap col
        float loc = sD[lane][0] + sD[lane][1];               // one_body+two_body
        sden[lane]  = d;
        s_d2[lane]  = d * d;
        s_num[lane] = loc * d * d;
    }
    __syncthreads();

    if (lane == 0) {
        float sd = 0.0f, sn = 0.0f;
        #pragma unroll
        for (int u = 0; u < NU; ++u) { sd += s_d2[u]; sn += s_num[u]; }
        energy_out[b] = sn / sd;
    }

    // --- scatter top_states = tanh(V + denom (x) feats) ---------------------
    if (winner[row] == b) {
        float* outp = vout + (size_t)row * (NU * FDIM);
        for (int u = 0; u < NU; ++u) {
            const float du = sden[u];
            for (int c = lane; c < FDIM; c += 32) {
                float t = A[u * FDIM + c] + du * sB[0][c];
                outp[u * FDIM + c] = tanhf(t);
            }
        }
    }
}

// ---------------------------------------------------------------------------
extern "C" void kernel_launch(void* const* d_in, const int* in_sizes, int n_in,
                              void* d_out, int out_size, void* d_ws, size_t ws_size,
                              hipStream_t stream) {
    const float* mo   = (const float*)d_in[0];
    const float* V_n  = (const float*)d_in[1];
    const int*   vidx = (const int*)  d_in[2];
    const float* h1   = (const float*)d_in[3];
    const int*   h1i  = (const int*)  d_in[4];
    const float* h2   = (const float*)d_in[5];
    const int*   h2i  = (const int*)  d_in[6];
    const float* gram = (const float*)d_in[7];

    const int B = in_sizes[2];                 // 8192
    const int E = in_sizes[4];                 // 65536
    const int F = in_sizes[0] / B;             // 128
    const int M = in_sizes[1] / (NU * F);      // 50000

    float* energy = (float*)d_out;
    float* vout   = (float*)d_out + B;

    // workspace layout
    float* agg1   = (float*)d_ws;                       // B*F
    float* agg2   = agg1 + (size_t)B * F;               // B*F
    int*   winner = (int*)(agg2 + (size_t)B * F);       // M

    const int nagg = B * F;
    {
        int n = nagg > M ? nagg : M;
        init_ws_kernel<<<(n + 255) / 256, 256, 0, stream>>>(agg1, agg2, winner, nagg, M);
    }
    {
        unsigned total = (unsigned)E * (unsigned)F;
        segsum_kernel<<<(total + 255) / 256, 256, 0, stream>>>(h1, h1i, h2, h2i,
                                                               agg1, agg2, total);
    }
    winner_kernel<<<(B + 255) / 256, 256, 0, stream>>>(vidx, winner, B);

    {
        size_t n4 = (size_t)M * NU * F / 4;
        copy_vn_kernel<<<8192, 256, 0, stream>>>((const float4*)V_n, (float4*)vout, n4);
    }

    elocal_main_kernel<<<B, 32, 0, stream>>>(mo, V_n, vidx, gram, agg1, agg2,
                                             winner, energy, vout);
}